// RGCNEncoder_12498354831444
// MI455X (gfx1250) — compile-verified
//
#include <hip/hip_runtime.h>

// ---------------------------------------------------------------------------
// RGCN encoder, MI455X (gfx1250).
// Reference resets hidden=embeddings every layer -> only the LAST layer's
// transform matters:  out = scatter_add(ht[rel] + bias) with
// ht[r] = H @ W[L-1, r]  (18 fp32 GEMMs of 100000x128x128), computed on the
// matrix cores with V_WMMA_F32_16X16X4_F32.
// ---------------------------------------------------------------------------

typedef __attribute__((ext_vector_type(2))) float v2f;
typedef __attribute__((ext_vector_type(4))) float v4f;
typedef __attribute__((ext_vector_type(8))) float v8f;

#define EDIM 128
// Words per K-pair row in LDS.  Each pair-row holds 128 cols x 2 interleaved
// K values (256 words) + 32 pad words.  288 % 64 == 32, so lanes 16-31
// (reading pair-row p+1) land on the disjoint half of the 64 banks ->
// conflict-free ds_load_b64 fragments.
#define PSTRIDE 288

// ---------------------------------------------------------------------------
// Phase 1: ht[r] = H @ W_r   (fp32 WMMA GEMM)
// Block: 256 threads (8 waves), 256-row node tile, blockIdx.y = relation.
// Wave w computes rows [32w, 32w+32) x all 128 cols: two A fragments,
// 16 v8f accumulators, 16 WMMAs per K-step against 8 B-fragment b64 loads.
// ---------------------------------------------------------------------------
__global__ __launch_bounds__(256) void rgcn_transform_kernel(
    const float* __restrict__ H,     // [nNodes][128]
    const float* __restrict__ Wl,    // [2R][128][128] (last layer)
    float*       __restrict__ ht,    // [chunkR][nNodes][128]
    int nNodes, int c0)
{
    __shared__ float Bs[64 * PSTRIDE];   // W_r, K-pair interleaved

    const int rloc = blockIdx.y;
    const float* Wr = Wl + (size_t)(c0 + rloc) * EDIM * EDIM;

    // Stage W_r: element (k, col) -> Bs[(k>>1)*PSTRIDE + 2*col + (k&1)].
    // A WMMA B-fragment (B[k][col], B[k+1][col]) is then one aligned b64 load.
    for (int i = threadIdx.x; i < EDIM * (EDIM / 4); i += 256) {
        const int k = i >> 5;            // 0..127
        const int q = (i & 31) << 2;     // col base, step 4
        const v4f w = *(const v4f*)&Wr[k * EDIM + q];
        float* d = &Bs[(k >> 1) * PSTRIDE + (k & 1)];
        d[2 * (q + 0)] = w.x;
        d[2 * (q + 1)] = w.y;
        d[2 * (q + 2)] = w.z;
        d[2 * (q + 3)] = w.w;
    }
    __syncthreads();

    const int lane = threadIdx.x & 31;
    const int l15  = lane & 15;
    const bool hi  = lane >= 16;             // lanes 16..31 carry K+2, K+3
    const int wid  = threadIdx.x >> 5;
    const int rowBase = blockIdx.x * 256 + wid * 32;

    // A-matrix rows for the two 16-row groups (clamped; masked at store).
    int m0 = rowBase + l15;        if (m0 >= nNodes) m0 = nNodes - 1;
    int m1 = rowBase + 16 + l15;   if (m1 >= nNodes) m1 = nNodes - 1;
    const float* __restrict__ arow0 = H + (size_t)m0 * EDIM;
    const float* __restrict__ arow1 = H + (size_t)m1 * EDIM;

    v8f acc0[8] = {};
    v8f acc1[8] = {};

    for (int kk = 0; kk < EDIM; kk += 4) {
        const int ka = kk + (hi ? 2 : 0);
        // A 16x4 fragments (ISA: v0 = K | K+2, v1 = K+1 | K+3); contiguous,
        // 8B-aligned -> global_load_b64 each.
        const v2f a0 = *(const v2f*)&arow0[ka];
        const v2f a1 = *(const v2f*)&arow1[ka];
        const float* __restrict__ bbase =
            &Bs[((kk >> 1) + (hi ? 1 : 0)) * PSTRIDE];
        #pragma unroll
        for (int c = 0; c < 8; ++c) {
            const v2f b = *(const v2f*)&bbase[(c * 16 + l15) * 2];  // 1x ds_load_b64
            acc0[c] = __builtin_amdgcn_wmma_f32_16x16x4_f32(
                false, a0, false, b, (short)0, acc0[c], false, false);
            acc1[c] = __builtin_amdgcn_wmma_f32_16x16x4_f32(
                false, a1, false, b, (short)0, acc1[c], false, false);
        }
    }

    // C/D layout: VGPR v -> M = v (lanes 0-15) / v+8 (lanes 16-31), N = 16c+l15
    float* __restrict__ outR = ht + (size_t)rloc * nNodes * EDIM;
    #pragma unroll
    for (int v = 0; v < 8; ++v) {
        const int row0 = rowBase + (hi ? v + 8 : v);
        if (row0 < nNodes) {
            float* __restrict__ orow = outR + (size_t)row0 * EDIM + l15;
            #pragma unroll
            for (int c = 0; c < 8; ++c) orow[c * 16] = acc0[c][v];
        }
        const int row1 = rowBase + 16 + (hi ? v + 8 : v);
        if (row1 < nNodes) {
            float* __restrict__ orow = outR + (size_t)row1 * EDIM + l15;
            #pragma unroll
            for (int c = 0; c < 8; ++c) orow[c * 16] = acc1[c][v];
        }
    }
}

// ---------------------------------------------------------------------------
// Phase 2: per-edge gather + bias + scatter-add (hardware f32 atomics, output
// is 51 MB -> L2-resident).  One wave per edge; lane owns 4 columns.
// Relation-range filter [c0,c1) supports ht chunking in workspace.
// ---------------------------------------------------------------------------
__global__ __launch_bounds__(256) void rgcn_edge_kernel(
    const int*   __restrict__ eidx,   // [2][nEdges]
    const int*   __restrict__ etype,  // [nEdges]
    const float* __restrict__ ht,     // [chunkR][nNodes][128]
    const float* __restrict__ bl,     // [2R][128] (last layer)
    float*       __restrict__ out,    // [nNodes][128]
    int nNodes, int nEdges, int c0, int c1, int nRel)
{
    const int lane = threadIdx.x & 31;
    const int wid  = threadIdx.x >> 5;
    const int e    = blockIdx.x * 8 + wid;
    if (e >= nEdges) return;

    const int src = eidx[e];
    const int dst = eidx[nEdges + e];
    const int rel = etype[e];
    const int col = lane * 4;

    // forward: out[dst] += ht[rel][src] + b[rel]
    if (rel >= c0 && rel < c1) {
        const v4f h = *(const v4f*)(ht + ((size_t)(rel - c0) * nNodes + src) * EDIM + col);
        const v4f b = *(const v4f*)(bl + (size_t)rel * EDIM + col);
        float* orow = out + (size_t)dst * EDIM + col;
        unsafeAtomicAdd(orow + 0, h.x + b.x);
        unsafeAtomicAdd(orow + 1, h.y + b.y);
        unsafeAtomicAdd(orow + 2, h.z + b.z);
        unsafeAtomicAdd(orow + 3, h.w + b.w);
    }
    // reverse: out[src] += ht[rel+R][dst] + b[rel+R]
    const int rb = rel + nRel;
    if (rb >= c0 && rb < c1) {
        const v4f h = *(const v4f*)(ht + ((size_t)(rb - c0) * nNodes + dst) * EDIM + col);
        const v4f b = *(const v4f*)(bl + (size_t)rb * EDIM + col);
        float* orow = out + (size_t)src * EDIM + col;
        unsafeAtomicAdd(orow + 0, h.x + b.x);
        unsafeAtomicAdd(orow + 1, h.y + b.y);
        unsafeAtomicAdd(orow + 2, h.z + b.z);
        unsafeAtomicAdd(orow + 3, h.w + b.w);
    }
}

__global__ void zero_f32_kernel(float* __restrict__ p, long n) {
    const long i = (long)blockIdx.x * blockDim.x + threadIdx.x;
    if (i < n) p[i] = 0.0f;
}

// ---------------------------------------------------------------------------
extern "C" void kernel_launch(void* const* d_in, const int* in_sizes, int n_in,
                              void* d_out, int out_size, void* d_ws, size_t ws_size,
                              hipStream_t stream) {
    const int*   edge_index = (const int*)  d_in[0];  // [2][NE]
    const int*   edge_type  = (const int*)  d_in[1];  // [NE]
    const float* emb        = (const float*)d_in[2];  // [N][128]
    const float* weights    = (const float*)d_in[3];  // [L][2R][128][128]
    const float* biases     = (const float*)d_in[4];  // [L][2R][128]
    float*       out        = (float*)d_out;

    const int E = EDIM, R = 9, L = 3, R2 = 2 * R;
    const int nNodes = in_sizes[2] / E;
    const int nEdges = in_sizes[1];

    // Only the last layer's weights/biases matter (hidden reset each layer).
    const float* Wl = weights + (size_t)(L - 1) * R2 * E * E;
    const float* bl = biases  + (size_t)(L - 1) * R2 * E;
    float* ht = (float*)d_ws;

    // Chunk relations so the ht slab fits the workspace.
    const size_t bytesPerRel = (size_t)nNodes * E * sizeof(float);
    int chunkR = (bytesPerRel > 0) ? (int)(ws_size / bytesPerRel) : R2;
    if (chunkR < 1)  chunkR = 1;
    if (chunkR > R2) chunkR = R2;

    // Zero the output accumulator.
    {
        const long n = (long)nNodes * E;
        const int blocks = (int)((n + 255) / 256);
        zero_f32_kernel<<<blocks, 256, 0, stream>>>(out, n);
    }

    const int mBlocks = (nNodes + 255) / 256;
    const int eBlocks = (nEdges + 7) / 8;

    for (int c0 = 0; c0 < R2; c0 += chunkR) {
        const int c1 = (c0 + chunkR < R2) ? (c0 + chunkR) : R2;
        dim3 grid(mBlocks, c1 - c0);
        rgcn_transform_kernel<<<grid, 256, 0, stream>>>(emb, Wl, ht, nNodes, c0);
        rgcn_edge_kernel<<<eBlocks, 256, 0, stream>>>(
            edge_index, edge_type, ht, bl, out, nNodes, nEdges, c0, c1, R);
    }
}